// Cluster_47614007444074
// MI455X (gfx1250) — compile-verified
//
#include <hip/hip_runtime.h>
#include <math.h>

// ---------------------------------------------------------------------------
// GMM e-step specialized: D=128, K=4. Only the class y[b] of each sample is
// needed, so we bucket samples by class and do per-class Cholesky + L^-1 in
// LDS, then WMMA (f32 16x16x4) batched quadratic forms.
// ---------------------------------------------------------------------------

typedef __attribute__((ext_vector_type(2))) float v2f;
typedef __attribute__((ext_vector_type(8))) float v8f;

#define DIM    128
#define KCOMP  4
#define LDA    129          // LDS row stride (floats) to avoid bank conflicts

__device__ __constant__ const float kLog2Pi  = 1.8378770664093453f;
__device__ __constant__ const float kNegLogK = -1.3862943611198906f; // -log(4)

// ---------------- bucketing kernels ----------------------------------------

__global__ void k_zero_counts(int* counts, int nC) {
    int i = blockIdx.x * blockDim.x + threadIdx.x;
    if (i < nC) counts[i] = 0;
}

__global__ void k_count(const int* __restrict__ y, int* counts, int nB) {
    int i = blockIdx.x * blockDim.x + threadIdx.x;
    if (i < nB) atomicAdd(&counts[y[i]], 1);
}

__global__ void k_scan(const int* __restrict__ counts, int* offsets, int* fill, int nC) {
    if (threadIdx.x == 0 && blockIdx.x == 0) {
        int acc = 0;
        for (int c = 0; c < nC; ++c) {
            offsets[c] = acc;
            fill[c]    = acc;
            acc += counts[c];
        }
        offsets[nC] = acc;
    }
}

__global__ void k_scatter(const int* __restrict__ y, int* fill, int* bucket, int nB) {
    int i = blockIdx.x * blockDim.x + threadIdx.x;
    if (i < nB) {
        int pos = atomicAdd(&fill[y[i]], 1);
        bucket[pos] = i;
    }
}

// ---------------- main per-class kernel ------------------------------------
// grid.x = C classes, block = 128 threads (4 waves of 32).

__global__ __launch_bounds__(128) void k_class(
        const float* __restrict__ x,          // [B,128]
        const float* __restrict__ mu,         // [C,4,128]
        const float* __restrict__ cov,        // [C,4,128,128]
        const int*   __restrict__ counts,
        const int*   __restrict__ offsets,
        const int*   __restrict__ bucket,
        float*       __restrict__ logp_ws)    // [B,4]
{
    __shared__ float A[DIM][LDA];     // Sigma -> L (lower) in place
    __shared__ float W[DIM][LDA];     // W = L^-1 (lower triangular, upper zeroed)
    __shared__ float Dt[16][LDA];     // 16-sample diff tile
    __shared__ float red[128];
    __shared__ float lds_maha[16];
    __shared__ float s_logdet;

    const int c    = blockIdx.x;
    const int tid  = threadIdx.x;
    const int lane = tid & 31;
    const int wv   = tid >> 5;        // wave id 0..3
    const int M    = lane & 15;       // row (A frag) / col (B frag) within tile
    const int hi   = lane >> 4;       // 0 or 1 -> K-pair select / C row split

    const int n   = counts[c];
    const int off = offsets[c];
    if (n == 0) return;               // uniform per block
    const int ntiles = (n + 15) >> 4;

    for (int k = 0; k < KCOMP; ++k) {
        const float* covck = cov + (size_t)(c * KCOMP + k) * DIM * DIM;
        const float* muck  = mu  + (size_t)(c * KCOMP + k) * DIM;

        // ---- load Sigma into LDS (coalesced) ----
        for (int i = tid; i < DIM * DIM; i += 128)
            A[i >> 7][i & 127] = covck[i];
        __syncthreads();

        // ---- Cholesky (right-looking), thread i owns row i ----
        for (int kk = 0; kk < DIM; ++kk) {
            if (tid == 0) A[kk][kk] = sqrtf(A[kk][kk]);
            __syncthreads();
            if (tid > kk) A[tid][kk] /= A[kk][kk];
            __syncthreads();
            if (tid > kk) {
                float aik = A[tid][kk];
                for (int j = kk + 1; j <= tid; ++j)
                    A[tid][j] -= aik * A[j][kk];
            }
            __syncthreads();
        }

        // ---- logdet = sum log diag(L) ----
        red[tid] = logf(A[tid][tid]);
        __syncthreads();
        if (tid == 0) {
            float s = 0.f;
            for (int i = 0; i < 128; ++i) s += red[i];
            s_logdet = s;
        }

        // ---- W = L^-1 : thread j solves column j by forward substitution ----
        {
            const int j = tid;
            for (int i = 0; i < j; ++i) W[i][j] = 0.0f;
            for (int i = j; i < DIM; ++i) {
                float s = (i == j) ? 1.0f : 0.0f;
                for (int kk2 = j; kk2 < i; ++kk2)
                    s -= A[i][kk2] * W[kk2][j];
                W[i][j] = s / A[i][i];
            }
        }
        __syncthreads();

        // ---- sample tiles: V = diff * W^T via WMMA f32 16x16x4, maha=rowsum(V^2)
        for (int t = 0; t < ntiles; ++t) {
            {   // load 16x128 diff tile; 128 thr = 16 rows x 8 col-groups of 16
                int r  = tid >> 3;
                int c0 = (tid & 7) << 4;
                int si = t * 16 + r;
                if (si < n) {
                    const float* xb = x + (size_t)bucket[off + si] * DIM;
                    for (int j = 0; j < 16; ++j)
                        Dt[r][c0 + j] = xb[c0 + j] - muck[c0 + j];
                } else {
                    for (int j = 0; j < 16; ++j) Dt[r][c0 + j] = 0.0f;
                }
            }
            if (tid < 16) lds_maha[tid] = 0.0f;
            __syncthreads();

            // each wave computes N-tiles {2*wv, 2*wv+1}; full EXEC for WMMA
            for (int nt = 2 * wv; nt <= 2 * wv + 1; ++nt) {
                const int n0 = nt * 16;
                v8f acc = {};
                for (int jk = 0; jk < DIM; jk += 4) {
                    // A frag (16x4 f32): lanes0-15 -> K0,K1 ; lanes16-31 -> K2,K3
                    v2f a, b;
                    a.x = Dt[M][jk + 2 * hi + 0];
                    a.y = Dt[M][jk + 2 * hi + 1];
                    // B frag (4x16): B[j][i] = W[i][j]; lane gives column i=n0+M
                    b.x = W[n0 + M][jk + 2 * hi + 0];
                    b.y = W[n0 + M][jk + 2 * hi + 1];
                    acc = __builtin_amdgcn_wmma_f32_16x16x4_f32(
                        false, a, false, b, (short)0, acc, false, false);
                }
                // C frag: vgpr r, lane -> row (r + 8*hi), col M. Reduce cols.
                for (int r = 0; r < 8; ++r) {
                    float v = acc[r] * acc[r];
                    v += __shfl_xor(v, 1);
                    v += __shfl_xor(v, 2);
                    v += __shfl_xor(v, 4);
                    v += __shfl_xor(v, 8);
                    if (M == 0) atomicAdd(&lds_maha[r + 8 * hi], v);
                }
            }
            __syncthreads();

            if (tid < 16) {
                int si = t * 16 + tid;
                if (si < n) {
                    int b = bucket[off + si];
                    float lp = -0.5f * (128.0f * kLog2Pi + lds_maha[tid]) - s_logdet;
                    logp_ws[b * KCOMP + k] = lp;
                }
            }
            __syncthreads();
        }
        __syncthreads();   // before next component overwrites A/W
    }
}

// ---------------- final per-sample kernel ----------------------------------

__global__ void k_final(const float* __restrict__ lp4, float* __restrict__ out, int nB) {
    int b = blockIdx.x * blockDim.x + threadIdx.x;
    if (b >= nB) return;
    float lp[KCOMP];
    for (int k = 0; k < KCOMP; ++k) lp[k] = lp4[b * KCOMP + k];

    float mn = lp[0];
    for (int k = 1; k < KCOMP; ++k) mn = fminf(mn, lp[k]);
    float sc[KCOMP]; float ssum = 0.f;
    for (int k = 0; k < KCOMP; ++k) { sc[k] = lp[k] - mn; ssum += sc[k]; }
    float inv = 1.0f / fabsf(ssum);

    float l2[KCOMP]; float mx = -INFINITY;
    for (int k = 0; k < KCOMP; ++k) { l2[k] = lp[k] + kNegLogK; mx = fmaxf(mx, l2[k]); }
    float se = 0.f;
    for (int k = 0; k < KCOMP; ++k) se += expf(l2[k] - mx);
    float lse = mx + logf(se);

    out[b * 9 + 0] = lse;
    for (int k = 0; k < KCOMP; ++k) out[b * 9 + 1 + k] = expf(l2[k] - lse);
    for (int k = 0; k < KCOMP; ++k) out[b * 9 + 5 + k] = sc[k] * inv - lse;
}

// ---------------- launcher --------------------------------------------------

extern "C" void kernel_launch(void* const* d_in, const int* in_sizes, int n_in,
                              void* d_out, int out_size, void* d_ws, size_t ws_size,
                              hipStream_t stream) {
    const float* x   = (const float*)d_in[0];
    const int*   y   = (const int*)  d_in[1];
    const float* mu  = (const float*)d_in[2];
    const float* cov = (const float*)d_in[3];
    float* out = (float*)d_out;

    const int nB = in_sizes[1];                       // 4096
    const int nC = in_sizes[2] / (KCOMP * DIM);       // 100

    char* ws = (char*)d_ws;
    int*   counts  = (int*)(ws + 0);
    int*   offsets = (int*)(ws + 1024);
    int*   fill    = (int*)(ws + 2048);
    int*   bucket  = (int*)(ws + 4096);
    float* logp_ws = (float*)(ws + 4096 + ((size_t)nB + 64) * sizeof(int));

    k_zero_counts<<<(nC + 255) / 256, 256, 0, stream>>>(counts, nC);
    k_count      <<<(nB + 255) / 256, 256, 0, stream>>>(y, counts, nB);
    k_scan       <<<1, 32, 0, stream>>>(counts, offsets, fill, nC);
    k_scatter    <<<(nB + 255) / 256, 256, 0, stream>>>(y, fill, bucket, nB);
    k_class      <<<nC, 128, 0, stream>>>(x, mu, cov, counts, offsets, bucket, logp_ws);
    k_final      <<<(nB + 255) / 256, 256, 0, stream>>>(logp_ws, out, nB);
}